// Net_86328842650101
// MI455X (gfx1250) — compile-verified
//
#include <hip/hip_runtime.h>
#include <math.h>
#include <stdint.h>

// ---------------------------------------------------------------------------
// Affine frame: 3x3 rotation (row-major m[0..8]) + translation (m[9..11]).
// 48 bytes, 16B aligned so global/LDS b128 transfers are legal.
// ---------------------------------------------------------------------------
struct alignas(16) Aff { float m[12]; };

#define NCHUNKS 4096   // chunks in level-0 scan
#define K2_THREADS 256
#define PER (NCHUNKS / K2_THREADS)  // 16 aggregates per scan thread

__device__ inline Aff aff_identity() {
    Aff A;
    A.m[0] = 1.f; A.m[1] = 0.f; A.m[2] = 0.f;
    A.m[3] = 0.f; A.m[4] = 1.f; A.m[5] = 0.f;
    A.m[6] = 0.f; A.m[7] = 0.f; A.m[8] = 1.f;
    A.m[9] = 0.f; A.m[10] = 0.f; A.m[11] = 0.f;
    return A;
}

// O = X ∘ Y  (apply Y first, then X):  O.R = X.R*Y.R ; O.t = X.R*Y.t + X.t
__device__ inline Aff compose(const Aff& X, const Aff& Y) {
    Aff O;
#pragma unroll
    for (int r = 0; r < 3; ++r) {
        float x0 = X.m[r * 3 + 0], x1 = X.m[r * 3 + 1], x2 = X.m[r * 3 + 2];
        O.m[r * 3 + 0] = x0 * Y.m[0] + x1 * Y.m[3] + x2 * Y.m[6];
        O.m[r * 3 + 1] = x0 * Y.m[1] + x1 * Y.m[4] + x2 * Y.m[7];
        O.m[r * 3 + 2] = x0 * Y.m[2] + x1 * Y.m[5] + x2 * Y.m[8];
        O.m[9 + r]     = x0 * Y.m[9] + x1 * Y.m[10] + x2 * Y.m[11] + X.m[9 + r];
    }
    return O;
}

// Per-step transform in closed form (the reference's normalizations cancel
// analytically because |t| == dis and u.y^2+u.z^2 == sin^2(angle)).
__device__ inline Aff make_step(float dis, float ang, float dhd) {
    float sa, ca, sd, cd;
    sincosf(ang, &sa, &ca);
    sincosf(dhd, &sd, &cd);
    Aff M;
    M.m[0] = ca;      M.m[1] = -sa;     M.m[2] = 0.f;
    M.m[3] = cd * sa; M.m[4] = cd * ca; M.m[5] = -sd;
    M.m[6] = sd * sa; M.m[7] = sd * ca; M.m[8] = cd;
    M.m[9]  = dis * ca;
    M.m[10] = dis * cd * sa;
    M.m[11] = dis * sd * sa;
    return M;
}

// Gram-Schmidt on rows (det stays +1 via cross product) — combats fp32 drift
// the same way the reference's per-step vector normalization does.
__device__ inline void renorm(Aff& A) {
    float r0x = A.m[0], r0y = A.m[1], r0z = A.m[2];
    float inv = rsqrtf(r0x * r0x + r0y * r0y + r0z * r0z);
    r0x *= inv; r0y *= inv; r0z *= inv;
    float r1x = A.m[3], r1y = A.m[4], r1z = A.m[5];
    float d = r1x * r0x + r1y * r0y + r1z * r0z;
    r1x -= d * r0x; r1y -= d * r0y; r1z -= d * r0z;
    inv = rsqrtf(r1x * r1x + r1y * r1y + r1z * r1z);
    r1x *= inv; r1y *= inv; r1z *= inv;
    float r2x = r0y * r1z - r0z * r1y;
    float r2y = r0z * r1x - r0x * r1z;
    float r2z = r0x * r1y - r0y * r1x;
    A.m[0] = r0x; A.m[1] = r0y; A.m[2] = r0z;
    A.m[3] = r1x; A.m[4] = r1y; A.m[5] = r1z;
    A.m[6] = r2x; A.m[7] = r2y; A.m[8] = r2z;
}

// Seed frame from the three seed atoms (matches reference's first iteration).
__device__ inline Aff frame0(const float* __restrict__ x) {
    float ax = x[0], ay = x[1], az = x[2];
    float bx = x[3], by = x[4], bz = x[5];
    float cx = x[6], cy = x[7], cz = x[8];
    float mx = cx - bx, my = cy - by, mz = cz - bz;
    float inv = rsqrtf(mx * mx + my * my + mz * mz);
    mx *= inv; my *= inv; mz *= inv;                       // mk_n
    float px = bx - ax, py = by - ay, pz = bz - az;        // mk_1
    float nx = py * mz - pz * my;
    float ny = pz * mx - px * mz;
    float nz = px * my - py * mx;                          // nk
    inv = rsqrtf(nx * nx + ny * ny + nz * nz);
    nx *= inv; ny *= inv; nz *= inv;                       // nk_n
    float qx = ny * mz - nz * my;
    float qy = nz * mx - nx * mz;
    float qz = nx * my - ny * mx;                          // nk_n x mk_n
    Aff F;  // columns = (mk_n, nk_mk_n, nk_n); origin = c
    F.m[0] = mx; F.m[1] = qx; F.m[2] = nx;
    F.m[3] = my; F.m[4] = qy; F.m[5] = ny;
    F.m[6] = mz; F.m[7] = qz; F.m[8] = nz;
    F.m[9] = cx; F.m[10] = cy; F.m[11] = cz;
    return F;
}

// ---------------------------------------------------------------------------
// K1: each thread serially composes its chunk of step transforms -> aggregate.
// ---------------------------------------------------------------------------
__global__ void k_chunk_agg(const float* __restrict__ dis,
                            const float* __restrict__ ang,
                            const float* __restrict__ dhd,
                            Aff* __restrict__ agg, int n, int chunk) {
    int c = blockIdx.x * blockDim.x + threadIdx.x;
    if (c >= NCHUNKS) return;
    int s = c * chunk;
    int e = min(n, s + chunk);
    Aff P = aff_identity();
    if (s < e) {
        // Pull the tail cacheline of this thread's ~100B window early
        // (global_prefetch_b8); the serial compose loop hides the latency.
        __builtin_prefetch(&dis[e - 1], 0, 0);
        __builtin_prefetch(&ang[e - 1], 0, 0);
        __builtin_prefetch(&dhd[e - 1], 0, 0);
    }
    for (int k = s; k < e; ++k) {
        P = compose(P, make_step(dis[k], ang[k], dhd[k]));
    }
    renorm(P);
    agg[c] = P;
}

// ---------------------------------------------------------------------------
// K2: single workgroup. Async-DMA all chunk aggregates (192 KB) into LDS via
// the CDNA5 async-to-LDS engine (ASYNCcnt), then hierarchical scan:
//   per-thread serial product over PER aggregates -> Kogge-Stone over 256
//   thread totals in LDS -> emit exclusive prefixes (seed frame folded in),
//   written back over agg[] (the global copy is dead once staged in LDS).
// ---------------------------------------------------------------------------
__global__ __launch_bounds__(K2_THREADS) void k_scan(Aff* __restrict__ agg,
                                                     const float* __restrict__ xyz0) {
    __shared__ Aff sAgg[NCHUNKS];      // 192 KB (WGP LDS is 320 KB)
    __shared__ Aff sThr[K2_THREADS];   // 12 KB
    const int j = threadIdx.x;

    // Stage aggregates: 4096 * 48B = 12288 b128 pieces, 48 per thread,
    // coalesced across lanes, issued on the async engine.
    {
        const unsigned lbase = (unsigned)(uintptr_t)(&sAgg[0]);
        const int pieces = NCHUNKS * 3;              // 16B pieces
        const int per_thread = pieces / K2_THREADS;  // 48
#pragma unroll 4
        for (int i = 0; i < per_thread; ++i) {
            unsigned off   = (unsigned)(i * K2_THREADS + j) * 16u;
            unsigned laddr = lbase + off;
            asm volatile("global_load_async_to_lds_b128 %0, %1, %2"
                         :: "v"(laddr), "v"(off), "s"(agg)
                         : "memory");
        }
        asm volatile("s_wait_asynccnt 0x0" ::: "memory");
    }
    __syncthreads();

    // Per-thread serial product of its PER aggregates.
    Aff T = sAgg[j * PER];
#pragma unroll
    for (int i = 1; i < PER; ++i) T = compose(T, sAgg[j * PER + i]);
    renorm(T);
    sThr[j] = T;
    __syncthreads();

    // Kogge-Stone inclusive scan over 256 thread totals (order-preserving).
    for (int off = 1; off < K2_THREADS; off <<= 1) {
        Aff v = sThr[j];
        if (j >= off) v = compose(sThr[j - off], v);
        __syncthreads();
        sThr[j] = v;
        __syncthreads();
    }

    // Exclusive prefix for this thread's first chunk, seed frame folded in.
    Aff R = frame0(xyz0);
    if (j > 0) R = compose(R, sThr[j - 1]);
    renorm(R);
#pragma unroll
    for (int i = 0; i < PER; ++i) {
        int c = j * PER + i;
        agg[c] = R;                      // frame at the start of chunk c
        R = compose(R, sAgg[c]);
        if ((i & 7) == 7) renorm(R);
    }
}

// ---------------------------------------------------------------------------
// K3: each thread replays its chunk on top of its exclusive-prefix frame and
// writes atom positions (translation of the running frame).
// ---------------------------------------------------------------------------
__global__ void k_emit(const float* __restrict__ dis,
                       const float* __restrict__ ang,
                       const float* __restrict__ dhd,
                       const Aff* __restrict__ pref,
                       const float* __restrict__ xyz0,
                       float* __restrict__ out, int n, int chunk) {
    int c = blockIdx.x * blockDim.x + threadIdx.x;
    if (c == 0) {
#pragma unroll
        for (int i = 0; i < 9; ++i) out[i] = xyz0[i];  // seed atoms
    }
    if (c >= NCHUNKS) return;
    int s = c * chunk;
    int e = min(n, s + chunk);
    if (s >= e) return;
    __builtin_prefetch(&dis[e - 1], 0, 0);
    __builtin_prefetch(&ang[e - 1], 0, 0);
    __builtin_prefetch(&dhd[e - 1], 0, 0);
    Aff P = pref[c];
    for (int k = s; k < e; ++k) {
        Aff Q = compose(P, make_step(dis[k], ang[k], dhd[k]));
        out[(3 + k) * 3 + 0] = Q.m[9];
        out[(3 + k) * 3 + 1] = Q.m[10];
        out[(3 + k) * 3 + 2] = Q.m[11];
        P = Q;
    }
}

// ---------------------------------------------------------------------------
extern "C" void kernel_launch(void* const* d_in, const int* in_sizes, int n_in,
                              void* d_out, int out_size, void* d_ws, size_t ws_size,
                              hipStream_t stream) {
    const float* dis  = (const float*)d_in[0];
    const float* ang  = (const float*)d_in[1];
    const float* dhd  = (const float*)d_in[2];
    const float* xyz0 = (const float*)d_in[3];
    float* out = (float*)d_out;
    const int n = in_sizes[0];
    const int chunk = (n + NCHUNKS - 1) / NCHUNKS;

    Aff* agg = (Aff*)d_ws;   // NCHUNKS * 48B = 192 KB of d_ws; K2 overwrites
                             // it in place with the exclusive prefix frames.

    dim3 blk(256);
    dim3 grd(NCHUNKS / 256);
    hipLaunchKernelGGL(k_chunk_agg, grd, blk, 0, stream, dis, ang, dhd, agg, n, chunk);
    hipLaunchKernelGGL(k_scan, dim3(1), dim3(K2_THREADS), 0, stream, agg, xyz0);
    hipLaunchKernelGGL(k_emit, grd, blk, 0, stream, dis, ang, dhd, agg, xyz0, out, n, chunk);
}